// SplinePoissonProcess_74053826118023
// MI455X (gfx1250) — compile-verified
//
#include <hip/hip_runtime.h>
#include <stdint.h>

#define D_DIM   256
#define K_KNOTS 1024
#define B_BATCH 32768

#define TILE_D  32
#define TILE_B  64
#define LSTRIDE 1028   // floats per LDS prefix row: 16B-aligned, bank = (4*row + m) % 64 -> <=2-way conflicts

// ---------------- workspace layout (bytes) ----------------
// prefix : float  [D][K]      @ 0          (1,048,576)
// ys     : float2 [D][K]      @ 1,048,576  (2,097,152)   (y, slope) d-major, for t_inv gathers
// ysT    : float2 [K][D]      @ 3,145,728  (2,097,152)   (y, slope) k-major, for integral gathers
// kT     : int    [B]         @ 5,242,880  (131,072)
// dT     : float  [B]         @ 5,373,952  (131,072)
// total: 5,505,024 bytes

__device__ __forceinline__ float softplus_f(float x) {
    // numerically stable: max(x,0) + log1p(exp(-|x|))  (matches jax.nn.softplus)
    return fmaxf(x, 0.0f) + log1pf(expf(-fabsf(x)));
}

// -------- kernel 1: per-dim table build (one block per d, 256 threads) --------
__global__ __launch_bounds__(256)
void build_tables(const float* __restrict__ h,
                  const float* __restrict__ kl,
                  float*  __restrict__ prefix,
                  float2* __restrict__ ys,
                  float2* __restrict__ ysT) {
    __shared__ float sy[K_KNOTS];
    __shared__ float ssum[256];
    const int d = blockIdx.x;
    const int t = threadIdx.x;

    // softplus of 4 consecutive knots (coalesced b128 load)
    const float4 h4 = ((const float4*)(h + (size_t)d * K_KNOTS))[t];
    sy[4 * t + 0] = softplus_f(h4.x);
    sy[4 * t + 1] = softplus_f(h4.y);
    sy[4 * t + 2] = softplus_f(h4.z);
    sy[4 * t + 3] = softplus_f(h4.w);
    __syncthreads();

    // per-segment slope & area for owned knots
    float area[4], slope[4], yloc[4];
    #pragma unroll
    for (int i = 0; i < 4; ++i) {
        const int k = 4 * t + i;
        const float yk = sy[k];
        yloc[i] = yk;
        if (k < K_KNOTS - 1) {
            const float dtk = kl[k + 1] - kl[k];
            const float sl  = (sy[k + 1] - yk) / dtk;
            slope[i] = sl;
            area[i]  = yk * dtk + 0.5f * sl * dtk * dtk;
        } else {
            slope[i] = 0.0f;   // padded seg_slopes
            area[i]  = 0.0f;   // unused segment
        }
    }
    ssum[t] = area[0] + area[1] + area[2] + area[3];
    __syncthreads();

    // inclusive Hillis-Steele scan over 256 per-thread sums
    #pragma unroll
    for (int off = 1; off < 256; off <<= 1) {
        const float v = (t >= off) ? ssum[t - off] : 0.0f;
        __syncthreads();
        ssum[t] += v;
        __syncthreads();
    }

    float run = (t > 0) ? ssum[t - 1] : 0.0f;   // exclusive offset
    #pragma unroll
    for (int i = 0; i < 4; ++i) {
        const int k = 4 * t + i;
        prefix[(size_t)d * K_KNOTS + k] = run;   // prefix[k] = sum of areas[0..k-1]
        run += area[i];
        float2 p;
        p.x = yloc[i];
        p.y = slope[i];
        ys [(size_t)d * K_KNOTS + k] = p;
        ysT[(size_t)k * D_DIM  + d] = p;
    }
}

// -------- kernel 2: per-batch searchsorted of T (shared across all dims) --------
__global__ __launch_bounds__(256)
void prep_T(const float* __restrict__ kl,
            const float* __restrict__ T,
            int*   __restrict__ kT,
            float* __restrict__ dT) {
    __shared__ float skl[K_KNOTS];
    const int t = threadIdx.x;
    #pragma unroll
    for (int i = t; i < K_KNOTS; i += 256) skl[i] = kl[i];
    __syncthreads();

    const int b = blockIdx.x * 256 + t;
    const float tv = T[b];
    int pos = 0;                               // skl[0] == 0 <= tv always
    #pragma unroll
    for (int step = K_KNOTS / 2; step >= 1; step >>= 1)
        if (skl[pos + step] <= tv) pos += step;
    kT[b] = pos;
    dT[b] = tv - skl[pos];
}

// quadratic segment inversion (matches reference's where/clip semantics)
__device__ __forceinline__ float invert_seg(float c, float hk, float sk, float xk) {
    float delta;
    if (fabsf(sk) < 1e-10f) {
        delta = c / hk;
    } else {
        const float disc = fmaxf(hk * hk + 2.0f * sk * c, 0.0f);
        delta = (-hk + sqrtf(disc)) / sk;
    }
    return xk + delta;
}

// -------- kernel 3: integral + inversion, LDS-staged prefix tile --------
__global__ __launch_bounds__(256)
void spline_main(const float*  __restrict__ prefix,
                 const float2* __restrict__ ys,
                 const float2* __restrict__ ysT,
                 const float*  __restrict__ kl,
                 const int*    __restrict__ kT,
                 const float*  __restrict__ dT,
                 const float*  __restrict__ u,
                 float* __restrict__ out_int,
                 float* __restrict__ out_inv) {
    extern __shared__ float smem[];            // TILE_D*LSTRIDE + K_KNOTS floats
    float* spref = smem;
    float* skl   = smem + TILE_D * LSTRIDE;

    const int t  = threadIdx.x;
    const int d0 = blockIdx.y * TILE_D;
    const int b0 = blockIdx.x * TILE_B;

    // --- CDNA5 async global->LDS staging (ASYNCcnt) ---
    {
        const uint32_t lbase = (uint32_t)(uintptr_t)spref;           // low 32 bits of flat = LDS offset
        const uint64_t gbase = (uint64_t)(uintptr_t)(prefix + (size_t)d0 * K_KNOTS);
        #pragma unroll 4
        for (int r = 0; r < TILE_D; ++r) {
            const uint32_t laddr = lbase + (uint32_t)(r * LSTRIDE * 4 + t * 16);
            const uint64_t gaddr = gbase + (uint64_t)(r * K_KNOTS * 4 + t * 16);
            asm volatile("global_load_async_to_lds_b128 %0, %1, off"
                         :: "v"(laddr), "v"(gaddr) : "memory");
        }
        const uint32_t laddr2 = (uint32_t)(uintptr_t)skl + (uint32_t)(t * 16);
        const uint64_t gaddr2 = (uint64_t)(uintptr_t)kl + (uint64_t)(t * 16);
        asm volatile("global_load_async_to_lds_b128 %0, %1, off"
                     :: "v"(laddr2), "v"(gaddr2) : "memory");
        asm volatile("s_wait_asynccnt 0" ::: "memory");
    }
    __syncthreads();

    const int   dl = t & (TILE_D - 1);         // lane's dim within tile: 32 consecutive d's per wave
    const int   rg = t >> 5;                   // 0..7: batch row group
    const int   dg = d0 + dl;
    const float* myrow = spref + dl * LSTRIDE;
    const float* ucol  = u + dg;
    float*       icol  = out_int + dg;
    float*       vcol  = out_inv + dg;
    const float2* ysrow = ys + (size_t)dg * K_KNOTS;

    // process 2 batch rows per iteration: two independent LDS-probe chains in flight
    #pragma unroll
    for (int it = 0; it < TILE_B / 16; ++it) {
        const int bA = b0 + it * 16 + rg;      // wave-uniform
        const int bB = bA + 8;

        const int   ktA = kT[bA];
        const int   ktB = kT[bB];
        const float dtA = dT[bA];
        const float dtB = dT[bB];
        const float uvA = ucol[(size_t)bA * D_DIM];     // coalesced 128B per wave
        const float uvB = ucol[(size_t)bB * D_DIM];

        // ----- integral: prefix from LDS, (y,slope) coalesced from transposed table -----
        const float  pkA  = myrow[ktA];
        const float  pkB  = myrow[ktB];
        const float2 hsTA = ysT[(size_t)ktA * D_DIM + dg];
        const float2 hsTB = ysT[(size_t)ktB * D_DIM + dg];
        icol[(size_t)bA * D_DIM] = pkA + hsTA.x * dtA + 0.5f * hsTA.y * dtA * dtA;
        icol[(size_t)bB * D_DIM] = pkB + hsTB.x * dtB + 0.5f * hsTB.y * dtB * dtB;

        // ----- inversion: two interleaved 10-step binary searches in LDS -----
        int posA = 0, posB = 0;
        #pragma unroll
        for (int step = K_KNOTS / 2; step >= 1; step >>= 1) {
            const float pA = myrow[posA + step];
            const float pB = myrow[posB + step];
            if (pA <= uvA) posA += step;
            if (pB <= uvB) posB += step;
        }

        const float  cA  = uvA - myrow[posA];
        const float  cB  = uvB - myrow[posB];
        const float2 hsA = ysrow[posA];        // single 8B gather (L2-resident)
        const float2 hsB = ysrow[posB];
        const float  xkA = skl[posA];
        const float  xkB = skl[posB];

        vcol[(size_t)bA * D_DIM] = invert_seg(cA, hsA.x, hsA.y, xkA);
        vcol[(size_t)bB * D_DIM] = invert_seg(cB, hsB.x, hsB.y, xkB);
    }
}

extern "C" void kernel_launch(void* const* d_in, const int* in_sizes, int n_in,
                              void* d_out, int out_size, void* d_ws, size_t ws_size,
                              hipStream_t stream) {
    const float* h  = (const float*)d_in[0];   // (D,K)
    const float* kl = (const float*)d_in[1];   // (K,)
    const float* T  = (const float*)d_in[2];   // (B,)
    const float* u  = (const float*)d_in[3];   // (B,D)
    float* out = (float*)d_out;                // integral (B,D) then t_inv (B,D)

    char* ws = (char*)d_ws;
    float*  prefix = (float*) (ws + 0);
    float2* ys     = (float2*)(ws + 1048576);
    float2* ysT    = (float2*)(ws + 3145728);
    int*    kT     = (int*)   (ws + 5242880);
    float*  dT     = (float*) (ws + 5373952);

    build_tables<<<D_DIM, 256, 0, stream>>>(h, kl, prefix, ys, ysT);
    prep_T<<<B_BATCH / 256, 256, 0, stream>>>(kl, T, kT, dT);

    const size_t shmem = (size_t)(TILE_D * LSTRIDE + K_KNOTS) * sizeof(float); // 135,680 B (fits 320KB WGP LDS, 2 WGs/WGP)
    dim3 grid(B_BATCH / TILE_B, D_DIM / TILE_D);
    spline_main<<<grid, 256, shmem, stream>>>(prefix, ys, ysT, kl, kT, dT, u,
                                              out, out + (size_t)B_BATCH * D_DIM);
}